// BiLSTM_crf_73040213836434
// MI455X (gfx1250) — compile-verified
//
#include <hip/hip_runtime.h>
#include <hip/hip_bf16.h>

typedef __attribute__((ext_vector_type(2))) float v2f;
typedef __attribute__((ext_vector_type(8))) float v8f;

#define B_ 256
#define S_ 512
#define I_ 100
#define G4_ 400
#define NB_ 19
#define BOS_ 17
#define EOS_ 18

__device__ __forceinline__ v8f wmma_f32_4(v2f a, v2f b, v8f c) {
  // D = A(16x4,f32) * B(4x16,f32) + C(16x16,f32)
  return __builtin_amdgcn_wmma_f32_16x16x4_f32(false, a, false, b, (short)0, c,
                                               false, false);
}

// ---------------------------------------------------------------------------
// K0: repack weights into fragment-paired layouts:
//   WP[pr][n][q] = W[n][2*pr+q]  -> one aligned b64 load per WMMA B fragment
// ---------------------------------------------------------------------------
__global__ __launch_bounds__(256) void prep_kernel(
    const float* __restrict__ Wih_f, const float* __restrict__ Whh_f,
    const float* __restrict__ Wih_b, const float* __restrict__ Whh_b,
    const float* __restrict__ W1, const float* __restrict__ W2,
    const float* __restrict__ b1, const float* __restrict__ b2,
    float* __restrict__ WihP_f, float* __restrict__ WhhP_f,
    float* __restrict__ WihP_b, float* __restrict__ WhhP_b,
    float* __restrict__ W1P, float* __restrict__ W2P, float* __restrict__ b1p,
    float* __restrict__ b2p) {
  int tid = blockIdx.x * blockDim.x + threadIdx.x;
  int stride = gridDim.x * blockDim.x;
  // Wih (400,100) -> WihP[50][400][2]
  for (int e = tid; e < 50 * 400 * 2; e += stride) {
    int pr = e / 800, rem = e % 800, n = rem >> 1, q = rem & 1;
    int k = 2 * pr + q;
    WihP_f[e] = Wih_f[n * I_ + k];
    WihP_b[e] = Wih_b[n * I_ + k];
  }
  // Whh (400,100) -> WhhP[50][448][2], gate-padded columns (4 gates x 112)
  for (int e = tid; e < 50 * 448 * 2; e += stride) {
    int pr = e / 896, rem = e % 896, np = rem >> 1, q = rem & 1;
    int gate = np / 112, j = np % 112;
    int k = 2 * pr + q;
    float vf = 0.f, vb = 0.f;
    if (j < I_) {
      vf = Whh_f[(gate * I_ + j) * I_ + k];
      vb = Whh_b[(gate * I_ + j) * I_ + k];
    }
    WhhP_f[e] = vf;
    WhhP_b[e] = vb;
  }
  // W1 (100,200) -> W1P[100][112][2], zero-padded cols
  for (int e = tid; e < 100 * 112 * 2; e += stride) {
    int pr = e / 224, rem = e % 224, n = rem >> 1, q = rem & 1;
    int k = 2 * pr + q;
    W1P[e] = (n < 100) ? W1[n * 200 + k] : 0.f;
  }
  // W2 (19,100) -> W2P[56][32][2], zero-padded
  for (int e = tid; e < 56 * 32 * 2; e += stride) {
    int pr = e / 64, rem = e % 64, n = rem >> 1, q = rem & 1;
    int k = 2 * pr + q;
    W2P[e] = (n < NB_ && k < 100) ? W2[n * 100 + k] : 0.f;
  }
  for (int e = tid; e < 112; e += stride) b1p[e] = (e < 100) ? b1[e] : 0.f;
  for (int e = tid; e < 32; e += stride) b2p[e] = (e < NB_) ? b2[e] : 0.f;
}

// ---------------------------------------------------------------------------
// K1: fused embedding gather + input projection (both directions)
// ---------------------------------------------------------------------------
__global__ __launch_bounds__(256) void input_proj_kernel(
    const int* __restrict__ x, const float* __restrict__ emb,
    const float* __restrict__ WihP_f, const float* __restrict__ WihP_b,
    const float* __restrict__ bih_f, const float* __restrict__ bhh_f,
    const float* __restrict__ bih_b, const float* __restrict__ bhh_b,
    float* __restrict__ pre_f, float* __restrict__ pre_b) {
  __shared__ float xs[16 * I_];
  __shared__ int idx[16];
  const int row0 = blockIdx.x * 16;
  if (threadIdx.x < 16) idx[threadIdx.x] = x[row0 + threadIdx.x];
  __syncthreads();
  for (int e = threadIdx.x; e < 16 * I_; e += 256) {
    int m = e / I_, k = e % I_;
    xs[e] = emb[(long)idx[m] * I_ + k];
  }
  __syncthreads();
  const int lane = threadIdx.x & 31, wave = threadIdx.x >> 5;
  const int lm = lane & 15, lh = lane >> 4;
  const v2f* xs2 = (const v2f*)xs;
  for (int jj = wave; jj < 50; jj += 8) {
    const int dir = jj / 25, nt = jj % 25;
    const v2f* WP = (const v2f*)(dir ? WihP_b : WihP_f);
    const float* bi = dir ? bih_b : bih_f;
    const float* bh = dir ? bhh_b : bhh_f;
    float* out = dir ? pre_b : pre_f;
    const int n = nt * 16 + lm;
    v8f acc = {};
    for (int k0 = 0; k0 < I_; k0 += 4) {
      int pr = (k0 >> 1) + lh;
      v2f a = xs2[lm * 50 + pr];
      v2f b = WP[pr * 400 + n];
      acc = wmma_f32_4(a, b, acc);
    }
    const float bias = bi[n] + bh[n];
#pragma unroll
    for (int r = 0; r < 8; ++r) {
      int m = r + 8 * lh;
      out[(long)(row0 + m) * G4_ + n] = acc[r] + bias;
    }
  }
}

// ---------------------------------------------------------------------------
// K2: LSTM recurrence. 32 blocks x 224 threads (7 waves).
// Wave w owns gate-column slice [16w,16w+16) of all 4 gates -> gates and cell
// state stay in registers; only h crosses waves via 7KB LDS per step.
// ---------------------------------------------------------------------------
__global__ __launch_bounds__(224) void lstm_kernel(
    const float* __restrict__ pre_f, const float* __restrict__ pre_b,
    const float* __restrict__ WhhP_f, const float* __restrict__ WhhP_b,
    float* __restrict__ hbuf) {
  __shared__ float hl[16 * 112];  // 7.2 KB, cols 100..111 never read
  const int dir = blockIdx.x >> 4;
  const int b0 = (blockIdx.x & 15) * 16;
  const float* pre = dir ? pre_b : pre_f;
  const v2f* WP = (const v2f*)(dir ? WhhP_b : WhhP_f);
  const int hofs = dir ? I_ : 0;
  for (int e = threadIdx.x; e < 16 * 112; e += 224) hl[e] = 0.f;
  const int lane = threadIdx.x & 31, wave = threadIdx.x >> 5;  // wave 0..6
  const int lm = lane & 15, lh = lane >> 4;
  const int j = wave * 16 + lm;  // gate column owned by this lane
  const v2f* hl2 = (const v2f*)hl;
  float creg[8];
#pragma unroll
  for (int r = 0; r < 8; ++r) creg[r] = 0.f;
  __syncthreads();
  for (int s = 0; s < S_; ++s) {
    const int t = dir ? (S_ - 1 - s) : s;
    // G = h(16x100) @ WhhP(100x448): 4 gate tiles per wave
    v8f acc0 = {}, acc1 = {}, acc2 = {}, acc3 = {};
    for (int k0 = 0; k0 < I_; k0 += 4) {
      int pr = (k0 >> 1) + lh;
      v2f a = hl2[lm * 56 + pr];
      const v2f* wrow = WP + pr * 448;
      acc0 = wmma_f32_4(a, wrow[j], acc0);
      acc1 = wmma_f32_4(a, wrow[112 + j], acc1);
      acc2 = wmma_f32_4(a, wrow[224 + j], acc2);
      acc3 = wmma_f32_4(a, wrow[336 + j], acc3);
    }
    __syncthreads();  // all hl reads done
    if (j < I_) {
#pragma unroll
      for (int r = 0; r < 8; ++r) {
        int m = r + 8 * lh;
        long prow = (long)(b0 + m) * S_ + t;
        const float* p = pre + prow * G4_;
        float ig = acc0[r] + p[j];
        float fg = acc1[r] + p[I_ + j];
        float gg = acc2[r] + p[2 * I_ + j];
        float og = acc3[r] + p[3 * I_ + j];
        float si = 1.f / (1.f + __expf(-ig));
        float sf = 1.f / (1.f + __expf(-fg));
        float so = 1.f / (1.f + __expf(-og));
        float c = sf * creg[r] + si * tanhf(gg);
        float h = so * tanhf(c);
        creg[r] = c;
        hl[m * 112 + j] = h;
        hbuf[prow * (2 * I_) + hofs + j] = h;
      }
    }
    __syncthreads();  // hl writes visible before next GEMM
  }
}

// ---------------------------------------------------------------------------
// K3: fused two-layer projection: em = (h @ W1^T + b1) @ W2^T + b2
// ---------------------------------------------------------------------------
__global__ __launch_bounds__(128) void proj_kernel(
    const float* __restrict__ hbuf, const float* __restrict__ W1P,
    const float* __restrict__ W2P, const float* __restrict__ b1p,
    const float* __restrict__ b2p, float* __restrict__ em) {
  __shared__ float t1[4][16 * 112];  // 28.7 KB
  const int lane = threadIdx.x & 31, wave = threadIdx.x >> 5;
  const int lm = lane & 15, lh = lane >> 4;
  const int rt = blockIdx.x * 4 + wave;
  const int row0 = rt * 16;
  const v2f* hb2 = (const v2f*)hbuf;
  const v2f* W1P2 = (const v2f*)W1P;
  const v2f* W2P2 = (const v2f*)W2P;
  // stage 1: (16x200) @ (200x112)
  for (int nt = 0; nt < 7; ++nt) {
    const int n = nt * 16 + lm;
    v8f acc = {};
    for (int k0 = 0; k0 < 200; k0 += 4) {
      int pr = (k0 >> 1) + lh;
      v2f a = hb2[(long)(row0 + lm) * 100 + pr];
      v2f b = W1P2[pr * 112 + n];
      acc = wmma_f32_4(a, b, acc);
    }
    const float bias = b1p[n];
#pragma unroll
    for (int r = 0; r < 8; ++r) t1[wave][(r + 8 * lh) * 112 + n] = acc[r] + bias;
  }
  // stage 2: (16x112) @ (112x32), valid N = 19
  const v2f* t12 = (const v2f*)t1[wave];
  for (int nt = 0; nt < 2; ++nt) {
    const int n = nt * 16 + lm;
    v8f acc = {};
    for (int k0 = 0; k0 < 112; k0 += 4) {
      int pr = (k0 >> 1) + lh;
      v2f a = t12[lm * 56 + pr];
      v2f b = W2P2[pr * 32 + n];
      acc = wmma_f32_4(a, b, acc);
    }
#pragma unroll
    for (int r = 0; r < 8; ++r) {
      int m = r + 8 * lh;
      if (n < NB_) em[(long)(row0 + m) * NB_ + n] = acc[r] + b2p[n];
    }
  }
}

// ---------------------------------------------------------------------------
// K4: CRF gold-path scores, one thread per batch element
// ---------------------------------------------------------------------------
__global__ __launch_bounds__(256) void crf_score_kernel(
    const float* __restrict__ em, const int* __restrict__ target,
    const float* __restrict__ T, float* __restrict__ score) {
  const int b = threadIdx.x;
  const int* tg = target + (long)b * S_;
  const float* eb = em + (long)b * S_ * NB_;
  int prev = tg[0];
  float e_sc = eb[prev];
  float t_sc = T[BOS_ * NB_ + prev];
  for (int t = 1; t < S_; ++t) {
    int cur = tg[t];
    e_sc += eb[(long)t * NB_ + cur];
    t_sc += T[prev * NB_ + cur];
    prev = cur;
  }
  t_sc += T[prev * NB_ + EOS_];
  score[b] = e_sc + t_sc;
}

// ---------------------------------------------------------------------------
// K5: CRF forward algorithm (partition), one block (1 wave) per batch element
// ---------------------------------------------------------------------------
__global__ __launch_bounds__(32) void crf_part_kernel(
    const float* __restrict__ em, const float* __restrict__ T,
    float* __restrict__ part) {
  __shared__ float Ts[NB_ * NB_];
  __shared__ float av[NB_];
  const int b = blockIdx.x;
  const int j = threadIdx.x;
  for (int e = j; e < NB_ * NB_; e += 32) Ts[e] = T[e];
  const float* eb = em + (long)b * S_ * NB_;
  if (j < NB_) av[j] = T[BOS_ * NB_ + j] + eb[j];
  __syncthreads();
  for (int t = 1; t < S_; ++t) {
    float na = 0.f;
    if (j < NB_) {
      float mx = -3.4e38f;
#pragma unroll
      for (int i = 0; i < NB_; ++i) mx = fmaxf(mx, av[i] + Ts[i * NB_ + j]);
      float s = 0.f;
#pragma unroll
      for (int i = 0; i < NB_; ++i) s += __expf(av[i] + Ts[i * NB_ + j] - mx);
      na = mx + __logf(s) + eb[(long)t * NB_ + j];
    }
    __syncthreads();
    if (j < NB_) av[j] = na;
    __syncthreads();
  }
  if (j == 0) {
    float mx = -3.4e38f;
    for (int i = 0; i < NB_; ++i) mx = fmaxf(mx, av[i] + Ts[i * NB_ + EOS_]);
    float s = 0.f;
    for (int i = 0; i < NB_; ++i) s += __expf(av[i] + Ts[i * NB_ + EOS_] - mx);
    part[b] = mx + __logf(s);
  }
}

// ---------------------------------------------------------------------------
// K6: final reduction -> scalar NLL
// ---------------------------------------------------------------------------
__global__ __launch_bounds__(256) void crf_reduce_kernel(
    const float* __restrict__ score, const float* __restrict__ part,
    float* __restrict__ out) {
  __shared__ float red[256];
  const int b = threadIdx.x;
  red[b] = score[b] - part[b];
  __syncthreads();
  for (int s = 128; s > 0; s >>= 1) {
    if (b < s) red[b] += red[b + s];
    __syncthreads();
  }
  if (b == 0) out[0] = -red[0] / (float)B_;
}

// ---------------------------------------------------------------------------
extern "C" void kernel_launch(void* const* d_in, const int* in_sizes, int n_in,
                              void* d_out, int out_size, void* d_ws,
                              size_t ws_size, hipStream_t stream) {
  const int* x = (const int*)d_in[0];
  const int* target = (const int*)d_in[1];
  const float* emb = (const float*)d_in[2];
  const float* Wih_f = (const float*)d_in[3];
  const float* Whh_f = (const float*)d_in[4];
  const float* bih_f = (const float*)d_in[5];
  const float* bhh_f = (const float*)d_in[6];
  const float* Wih_b = (const float*)d_in[7];
  const float* Whh_b = (const float*)d_in[8];
  const float* bih_b = (const float*)d_in[9];
  const float* bhh_b = (const float*)d_in[10];
  const float* W1 = (const float*)d_in[11];
  const float* b1 = (const float*)d_in[12];
  const float* W2 = (const float*)d_in[13];
  const float* b2 = (const float*)d_in[14];
  const float* T = (const float*)d_in[15];
  float* out = (float*)d_out;

  const long ROWS = (long)B_ * S_;  // 131072
  float* ws = (float*)d_ws;
  size_t o = 0;
  float* pre_f = ws + o;  o += (size_t)ROWS * G4_;
  float* pre_b = ws + o;  o += (size_t)ROWS * G4_;
  float* hbuf = ws + o;   o += (size_t)ROWS * 2 * I_;
  float* em = ws + o;     o += (size_t)ROWS * NB_;
  float* WihP_f = ws + o; o += 50 * 400 * 2;
  float* WhhP_f = ws + o; o += 50 * 448 * 2;
  float* WihP_b = ws + o; o += 50 * 400 * 2;
  float* WhhP_b = ws + o; o += 50 * 448 * 2;
  float* W1P = ws + o;    o += 100 * 112 * 2;
  float* W2P = ws + o;    o += 56 * 32 * 2;
  float* b1p = ws + o;    o += 112;
  float* b2p = ws + o;    o += 32;
  float* score = ws + o;  o += B_;
  float* part = ws + o;   o += B_;

  prep_kernel<<<160, 256, 0, stream>>>(Wih_f, Whh_f, Wih_b, Whh_b, W1, W2, b1,
                                       b2, WihP_f, WhhP_f, WihP_b, WhhP_b, W1P,
                                       W2P, b1p, b2p);
  input_proj_kernel<<<(int)(ROWS / 16), 256, 0, stream>>>(
      x, emb, WihP_f, WihP_b, bih_f, bhh_f, bih_b, bhh_b, pre_f, pre_b);
  lstm_kernel<<<32, 224, 0, stream>>>(pre_f, pre_b, WhhP_f, WhhP_b, hbuf);
  proj_kernel<<<(int)(ROWS / 64), 128, 0, stream>>>(hbuf, W1P, W2P, b1p, b2p,
                                                    em);
  crf_score_kernel<<<1, 256, 0, stream>>>(em, target, T, score);
  crf_part_kernel<<<B_, 32, 0, stream>>>(em, T, part);
  crf_reduce_kernel<<<1, 256, 0, stream>>>(score, part, out);
}